// SLDS_62921270886611
// MI455X (gfx1250) — compile-verified
//
#include <hip/hip_runtime.h>
#include <math.h>

#define KC 16    // components
#define DD 32    // hidden dim
#define OO 16    // obs dim
#define TT 1024  // sequence length
#define LOG2PI 1.8378770664093453f

typedef float v2f __attribute__((ext_vector_type(2)));
typedef float v8f __attribute__((ext_vector_type(8)));

// ---- WMMA f32 16x16x4 helpers (wave32 fragment layouts per CDNA5 ISA 7.12.2) ----
static __device__ __forceinline__ v8f wmma4(v2f a, v2f b, v8f c) {
  return __builtin_amdgcn_wmma_f32_16x16x4_f32(false, a, false, b, (short)0, c, false, false);
}
// A fragment, element (m,k) = M[m][k]  (row-major source)
static __device__ __forceinline__ v2f frag_a_rm(const float* M, int ld, int m0, int k0, int lane) {
  int m = m0 + (lane & 15);
  int kb = k0 + ((lane >> 4) << 1);
  v2f a; a.x = M[m * ld + kb]; a.y = M[m * ld + kb + 1]; return a;
}
// A fragment of M^T, element (m,k) = M[k][m]  (row-contiguous read -> no bank conflicts)
static __device__ __forceinline__ v2f frag_a_tr(const float* M, int ld, int m0, int k0, int lane) {
  int m = m0 + (lane & 15);
  int kb = k0 + ((lane >> 4) << 1);
  v2f a; a.x = M[kb * ld + m]; a.y = M[(kb + 1) * ld + m]; return a;
}
// B fragment, element (k,n) = M[k][n]
static __device__ __forceinline__ v2f frag_b_rm(const float* M, int ld, int k0, int n0, int lane) {
  int n = n0 + (lane & 15);
  int kb = k0 + ((lane >> 4) << 1);
  v2f b; b.x = M[kb * ld + n]; b.y = M[(kb + 1) * ld + n]; return b;
}
static __device__ __forceinline__ void store_d_tile(float* M, int ld, int m0, int n0, int lane, v8f d) {
  int n = n0 + (lane & 15);
  int mb = m0 + ((lane >> 4) << 3);
#pragma unroll
  for (int r = 0; r < 8; ++r) M[(mb + r) * ld + n] = d[r];
}
static __device__ __forceinline__ v8f load_c_tile(const float* M, int ld, int m0, int n0, int lane) {
  int n = n0 + (lane & 15);
  int mb = m0 + ((lane >> 4) << 3);
  v8f c;
#pragma unroll
  for (int r = 0; r < 8; ++r) c[r] = M[(mb + r) * ld + n];
  return c;
}

__global__ __launch_bounds__(256) void slds_loglik_kernel(
    const float* __restrict__ data,   // (B,T,OO)
    const float* __restrict__ tlg,    // (K,K) transition logits
    const float* __restrict__ A,      // (K,D,D)
    const float* __restrict__ ltn,    // (D,)
    const float* __restrict__ Cg,     // (D,OO)
    const float* __restrict__ lon,    // (OO,)
    float* __restrict__ out) {        // (B,)
  // ---- persistent state ----
  __shared__ float sP[KC * DD * DD];      // covariances, row-major per k (64KB)
  __shared__ float sM[KC * DD];           // posterior means
  __shared__ float sMnew[KC * DD];        // predicted means
  __shared__ float sMb[KC * DD];          // mbar
  __shared__ float sW[KC], sWnew[KC];
  __shared__ float sLogT[KC * KC];        // row-normalized log transition
  __shared__ float sPi[KC * 17];          // responsibilities (padded)
  __shared__ float sC[DD * OO];           // observation matrix
  __shared__ float sQ[DD], sR[OO];
  __shared__ float sV[KC * OO];           // innovations
  __shared__ float sU[KC * OO];           // u = L^-1 v
  // ---- phase-shared scratch (64KB): predict: Mbar[j] 32x32 tiles (H in place)
  //      obs: PCT (16x33 per k) at [0..8448) + S/L (16x17 per k) at [8448..12800)
  __shared__ float sScratch[KC * DD * DD];

  const int tid = threadIdx.x;
  const int lane = tid & 31;
  const int wave = tid >> 5;
  const float* ybase = data + (size_t)blockIdx.x * TT * OO;

  // ================= init =================
  {
    int i = tid >> 4, j = tid & 15;
    sLogT[i * KC + j] = tlg[i * KC + j];
  }
  __syncthreads();
  {
    int i = tid >> 4, j = tid & 15;
    float mx = -INFINITY;
    for (int c = 0; c < KC; ++c) mx = fmaxf(mx, sLogT[i * KC + c]);
    float s = 0.f;
    for (int c = 0; c < KC; ++c) s += __expf(sLogT[i * KC + c] - mx);
    float nv = sLogT[i * KC + j] - (mx + __logf(s));
    __syncthreads();
    sLogT[i * KC + j] = nv;
  }
  for (int idx = tid; idx < DD * OO; idx += 256) sC[idx] = Cg[idx];
  if (tid < DD) sQ[tid] = __expf(ltn[tid]);
  if (tid < OO) sR[tid] = __expf(lon[tid]);
  for (int idx = tid; idx < KC * DD; idx += 256) sMnew[idx] = 0.f;
  for (int idx = tid; idx < KC * DD * DD; idx += 256) {
    int rem = idx & (DD * DD - 1);
    sP[idx] = ((rem >> 5) == (rem & 31)) ? 1.f : 0.f;
  }
  __syncthreads();
  if (tid < KC) sWnew[tid] = sLogT[tid];  // logT[0][k]
  __syncthreads();

  // ================= time loop =================
  for (int t = 0; t < TT; ++t) {
    if (t > 0) {
      // ---- S1: gating pi, wnew, mbar (wave-local per j-group of 16 threads) ----
      {
        int j = tid >> 4, i = tid & 15;
        sPi[j * 17 + i] = sW[i] + sLogT[i * KC + j];
        __builtin_amdgcn_wave_barrier();
        float mx = -INFINITY;
        for (int c = 0; c < KC; ++c) mx = fmaxf(mx, sPi[j * 17 + c]);
        float s = 0.f;
        for (int c = 0; c < KC; ++c) s += __expf(sPi[j * 17 + c] - mx);
        float wn = mx + __logf(s);
        float p = __expf(sPi[j * 17 + i] - wn);
        __builtin_amdgcn_wave_barrier();
        sPi[j * 17 + i] = p;
        if (i == 0) sWnew[j] = wn;
        __builtin_amdgcn_wave_barrier();
        float a0 = 0.f, a1 = 0.f;
        for (int c = 0; c < KC; ++c) {
          float pc = sPi[j * 17 + c];
          a0 += pc * sM[c * DD + i];
          a1 += pc * sM[c * DD + i + 16];
        }
        sMb[j * DD + i] = a0;
        sMb[j * DD + i + 16] = a1;
      }
      __syncthreads();  // (#1) pi, mbar visible to all waves

      // ---- S2: Mbar[j] = sum_i pi[j,i]*(P[i] + m_i^T m_i) via WMMA ----
      // output (16 j-rows) x (16-wide chunk of flattened ab); 64 tiles, 8 per wave
      for (int tt2 = 0; tt2 < 8; ++tt2) {
        int ab0 = (wave * 8 + tt2) * 16;
        int ab = ab0 + (lane & 15);
        int aa = ab >> 5, bb = ab & 31;
        v8f acc = {};
#pragma unroll
        for (int kk = 0; kk < 4; ++kk) {
          v2f a = frag_a_rm(sPi, 17, 0, kk * 4, lane);  // pi[j][i]
          int i0 = kk * 4 + ((lane >> 4) << 1);
          v2f b;  // second moment fragment built on the fly
          b.x = sP[i0 * 1024 + ab] + sM[i0 * DD + aa] * sM[i0 * DD + bb];
          b.y = sP[(i0 + 1) * 1024 + ab] + sM[(i0 + 1) * DD + aa] * sM[(i0 + 1) * DD + bb];
          acc = wmma4(a, b, acc);
        }
        store_d_tile(sScratch, 1024, 0, ab0, lane, acc);  // Mbar[j][ab]
      }
      // ---- S3: mnew[j] = mbar[j] @ A[j] (VALU) ----
      {
        int j = tid >> 4, i = tid & 15;
        const float* Aj = A + j * DD * DD;
        float a0 = 0.f, a1 = 0.f;
        for (int d2 = 0; d2 < DD; ++d2) {
          float mb = sMb[j * DD + d2];
          a0 += mb * Aj[d2 * DD + i];
          a1 += mb * Aj[d2 * DD + i + 16];
        }
        sMnew[j * DD + i] = a0;
        sMnew[j * DD + i + 16] = a1;
      }
      __syncthreads();  // (#2) Mbar + mnew complete

      // ---- S4: per wave j in {w, w+8}: P[j] = A^T Mbar A + Q - mnew mnew^T ----
      for (int jj = 0; jj < 2; ++jj) {
        int j = wave + jj * 8;
        float* Mb = sScratch + j * 1024;  // Mbar[j] (symmetric), then H in place
        const float* Aj = A + j * DD * DD;
        v8f h[4];
#pragma unroll
        for (int ti = 0; ti < 2; ++ti)
#pragma unroll
          for (int tj = 0; tj < 2; ++tj) {
            v8f acc = {};
#pragma unroll
            for (int kk = 0; kk < 8; ++kk) {
              v2f a = frag_a_tr(Mb, 32, ti * 16, kk * 4, lane);  // Mbar (symmetry -> row read)
              v2f b = frag_b_rm(Aj, 32, kk * 4, tj * 16, lane);  // A[j]
              acc = wmma4(a, b, acc);
            }
            h[ti * 2 + tj] = acc;
          }
        __builtin_amdgcn_wave_barrier();
        store_d_tile(Mb, 32, 0, 0, lane, h[0]);
        store_d_tile(Mb, 32, 0, 16, lane, h[1]);
        store_d_tile(Mb, 32, 16, 0, lane, h[2]);
        store_d_tile(Mb, 32, 16, 16, lane, h[3]);
        __builtin_amdgcn_wave_barrier();
        float* Pj = sP + j * 1024;
        const float* mn = sMnew + j * DD;
#pragma unroll
        for (int ti = 0; ti < 2; ++ti)
#pragma unroll
          for (int tj = 0; tj < 2; ++tj) {
            v8f acc = {};
#pragma unroll
            for (int kk = 0; kk < 8; ++kk) {
              v2f a = frag_a_tr(Aj, 32, ti * 16, kk * 4, lane);  // A^T
              v2f b = frag_b_rm(Mb, 32, kk * 4, tj * 16, lane);  // H
              acc = wmma4(a, b, acc);
            }
            int n = tj * 16 + (lane & 15);
            int mb0 = ti * 16 + ((lane >> 4) << 3);
            float mnn = mn[n];
#pragma unroll
            for (int r = 0; r < 8; ++r) {
              int mrow = mb0 + r;
              float val = acc[r] - mn[mrow] * mnn;
              if (mrow == n) val += sQ[n];
              Pj[mrow * 32 + n] = val;
            }
          }
      }
      __syncthreads();  // (#3) predicted state ready
    }

    // ================= observation update (wave w owns k=w and k=w+8) =================
    const float* yt = ybase + t * OO;
    {
      // innovation v[k] = y - mnew[k] @ C
      int k = wave + ((lane >> 4) << 3);
      int o = lane & 15;
      const float* mrow = sMnew + k * DD;
      float pred = 0.f;
      for (int d2 = 0; d2 < DD; ++d2) pred += mrow[d2] * sC[d2 * OO + o];
      sV[k * OO + o] = yt[o] - pred;
    }
    // PCT[k] = C^T @ P[k]  (16x32), WMMA
    for (int jj = 0; jj < 2; ++jj) {
      int k = wave + jj * 8;
      const float* Pk = sP + k * 1024;
      float* PCTk = sScratch + k * 528;
#pragma unroll
      for (int tj = 0; tj < 2; ++tj) {
        v8f acc = {};
#pragma unroll
        for (int kk = 0; kk < 8; ++kk) {
          v2f a = frag_a_tr(sC, OO, 0, kk * 4, lane);      // C^T
          v2f b = frag_b_rm(Pk, 32, kk * 4, tj * 16, lane);
          acc = wmma4(a, b, acc);
        }
        store_d_tile(PCTk, 33, 0, tj * 16, lane, acc);
      }
    }
    __builtin_amdgcn_wave_barrier();
    // S[k] = PCT[k] @ C + diag(R)
    for (int jj = 0; jj < 2; ++jj) {
      int k = wave + jj * 8;
      const float* PCTk = sScratch + k * 528;
      float* Sk = sScratch + 8448 + k * 272;
      v8f acc = {};
#pragma unroll
      for (int kk = 0; kk < 8; ++kk) {
        v2f a = frag_a_rm(PCTk, 33, 0, kk * 4, lane);
        v2f b = frag_b_rm(sC, OO, kk * 4, 0, lane);
        acc = wmma4(a, b, acc);
      }
      int n = lane & 15;
      int mb0 = (lane >> 4) << 3;
#pragma unroll
      for (int r = 0; r < 8; ++r) {
        float val = acc[r];
        if (mb0 + r == n) val += sR[n];
        Sk[(mb0 + r) * 17 + n] = val;
      }
    }
    __builtin_amdgcn_wave_barrier();
    // Cholesky S = L L^T in place (16 lanes per component, wave-synchronous)
    {
      int k = wave + ((lane >> 4) << 3);
      int ig = lane & 15;
      float* Sk = sScratch + 8448 + k * 272;
      for (int jc = 0; jc < 16; ++jc) {
        if (ig == jc) {
          float s2 = Sk[jc * 17 + jc];
          for (int p = 0; p < jc; ++p) { float l = Sk[jc * 17 + p]; s2 -= l * l; }
          Sk[jc * 17 + jc] = sqrtf(fmaxf(s2, 1e-20f));
        }
        __builtin_amdgcn_wave_barrier();
        float ljj = Sk[jc * 17 + jc];
        if (ig > jc) {
          float s2 = Sk[ig * 17 + jc];
          for (int p = 0; p < jc; ++p) s2 -= Sk[ig * 17 + p] * Sk[jc * 17 + p];
          Sk[ig * 17 + jc] = s2 / ljj;
        }
        __builtin_amdgcn_wave_barrier();
      }
    }
    // forward solves: W = L^-1 PCT in place (2 RHS/lane), u = L^-1 v, loglik
    {
      int k = wave + ((lane >> 4) << 3);
      int local = lane & 15;
      const float* Lk = sScratch + 8448 + k * 272;
      float* Wk = sScratch + k * 528;
      for (int which = 0; which < 2; ++which) {
        int d2 = local + which * 16;
        float z[16];
#pragma unroll
        for (int o = 0; o < 16; ++o) {
          float s2 = Wk[o * 33 + d2];
          for (int p = 0; p < o; ++p) s2 -= Lk[o * 17 + p] * z[p];
          z[o] = s2 / Lk[o * 17 + o];
        }
#pragma unroll
        for (int o = 0; o < 16; ++o) Wk[o * 33 + d2] = z[o];
      }
      if (local == 0) {
        float z[16];
        float q2 = 0.f, ld2 = 0.f;
#pragma unroll
        for (int o = 0; o < 16; ++o) {
          float s2 = sV[k * OO + o];
          for (int p = 0; p < o; ++p) s2 -= Lk[o * 17 + p] * z[p];
          z[o] = s2 / Lk[o * 17 + o];
          sU[k * OO + o] = z[o];
          q2 += z[o] * z[o];
          ld2 += __logf(Lk[o * 17 + o]);
        }
        sW[k] = sWnew[k] + (-0.5f * q2 - ld2 - 0.5f * (float)OO * LOG2PI);
      }
    }
    __builtin_amdgcn_wave_barrier();
    // m[k] = mnew[k] + W^T u ;  P[k] -= W^T W  (WMMA)
    for (int jj = 0; jj < 2; ++jj) {
      int k = wave + jj * 8;
      const float* Wk = sScratch + k * 528;
      const float* uk = sU + k * OO;
      float acc0 = sMnew[k * DD + lane];
#pragma unroll
      for (int o = 0; o < 16; ++o) acc0 += Wk[o * 33 + lane] * uk[o];
      sM[k * DD + lane] = acc0;

      float* Pk = sP + k * 1024;
#pragma unroll
      for (int ti = 0; ti < 2; ++ti)
#pragma unroll
        for (int tj = 0; tj < 2; ++tj) {
          v8f c = load_c_tile(Pk, 32, ti * 16, tj * 16, lane);
#pragma unroll
          for (int r = 0; r < 8; ++r) c[r] = -c[r];
#pragma unroll
          for (int kk = 0; kk < 4; ++kk) {
            v2f a = frag_a_tr(Wk, 33, ti * 16, kk * 4, lane);  // W^T
            v2f b = frag_b_rm(Wk, 33, kk * 4, tj * 16, lane);  // W
            c = wmma4(a, b, c);                                // c = W^T W - P
          }
          int n = tj * 16 + (lane & 15);
          int mb0 = ti * 16 + ((lane >> 4) << 3);
#pragma unroll
          for (int r = 0; r < 8; ++r) Pk[(mb0 + r) * 32 + n] = -c[r];
        }
    }
    __syncthreads();  // (#4) end of step
  }

  if (tid == 0) {
    float mx = -INFINITY;
    for (int k = 0; k < KC; ++k) mx = fmaxf(mx, sW[k]);
    float s = 0.f;
    for (int k = 0; k < KC; ++k) s += __expf(sW[k] - mx);
    out[blockIdx.x] = mx + __logf(s);
  }
}

extern "C" void kernel_launch(void* const* d_in, const int* in_sizes, int n_in,
                              void* d_out, int out_size, void* d_ws, size_t ws_size,
                              hipStream_t stream) {
  (void)n_in; (void)out_size; (void)d_ws; (void)ws_size;
  const float* data = (const float*)d_in[0];
  const float* tlg  = (const float*)d_in[1];
  const float* A    = (const float*)d_in[2];
  const float* ltn  = (const float*)d_in[3];
  const float* Cg   = (const float*)d_in[4];
  const float* lon  = (const float*)d_in[5];
  float* out = (float*)d_out;
  int B = in_sizes[0] / (TT * OO);  // 16
  slds_loglik_kernel<<<dim3(B), dim3(256), 0, stream>>>(data, tlg, A, ltn, Cg, lon, out);
}